// SpatialAttention_44693429682689
// MI455X (gfx1250) — compile-verified
//
#include <hip/hip_runtime.h>
#include <math.h>

typedef __attribute__((ext_vector_type(2))) float v2f;
typedef __attribute__((ext_vector_type(4))) float v4f;
typedef __attribute__((ext_vector_type(8))) float v8f;

#define BATCH 16
#define HDIM 224
#define WDIM 224
#define CDIM 256
#define NPIX (BATCH * HDIM * WDIM)          // 802816
#define NELEM ((size_t)NPIX * CDIM)         // 205520896
#define GROUPS_PER_ROW (WDIM / 16)          // 14
#define NGROUPS (BATCH * HDIM * GROUPS_PER_ROW) // 50176

// ---------------------------------------------------------------------------
// Pass 1: per-pixel channel max + mean over 256 channels.
// One wave32 per pixel; each lane loads 2 x float4 (8 channels), then a
// 5-step shfl_xor tree reduction. x is streamed once -> non-temporal loads.
// ---------------------------------------------------------------------------
__global__ void pool_kernel(const float* __restrict__ x,
                            float2* __restrict__ pool) {
    int wid  = (blockIdx.x * blockDim.x + threadIdx.x) >> 5; // pixel id
    int lane = threadIdx.x & 31;
    if (wid >= NPIX) return;

    const v4f* px = (const v4f*)(x + (size_t)wid * CDIM);
    v4f a = __builtin_nontemporal_load(px + lane);        // ch [4*lane .. +3]
    v4f b = __builtin_nontemporal_load(px + lane + 32);   // ch [128+4*lane ..]

    float m = fmaxf(fmaxf(fmaxf(a.x, a.y), fmaxf(a.z, a.w)),
                    fmaxf(fmaxf(b.x, b.y), fmaxf(b.z, b.w)));
    float s = (a.x + a.y) + (a.z + a.w) + (b.x + b.y) + (b.z + b.w);

    #pragma unroll
    for (int off = 16; off >= 1; off >>= 1) {
        m = fmaxf(m, __shfl_xor(m, off));
        s += __shfl_xor(s, off);
    }
    if (lane == 0) {
        pool[wid] = make_float2(m, s * (1.0f / (float)CDIM));
    }
}

// ---------------------------------------------------------------------------
// Pass 2: 7x7x2->1 conv + sigmoid via V_WMMA_F32_16X16X4_F32.
//   attn[1x16] = W[1x98] x im2col[98x16], K padded to 100, 25 WMMAs.
// A-matrix 16x4 f32 layout (ISA 7.12.2): lanes 0-15 hold K=0,1 in v0,v1;
// lanes 16-31 hold K=2,3. We populate only row M=0 (lane 0 and lane 16);
// all other rows are zero, so only C row 0 is meaningful.
// B-matrix 4x16: VGPR0/1 = K {+0,+1}, lane group selects K {+0,+2}; since
// k = (dy*7+dx)*2 + c and k0 is even, each lane's B pair is exactly the
// (max, avg) float2 at one spatial tap -> single b64 load.
// ---------------------------------------------------------------------------
__global__ void conv_wmma_kernel(const float2* __restrict__ pool,
                                 const float* __restrict__ conv_w,
                                 const float* __restrict__ conv_b,
                                 float* __restrict__ attn) {
    __shared__ float wlds[104]; // 98 weights, zero-padded to K=100 (+slack)
    for (int t = threadIdx.x; t < 104; t += blockDim.x)
        wlds[t] = (t < 98) ? conv_w[t] : 0.0f;
    __syncthreads();

    int group = blockIdx.x * (blockDim.x >> 5) + (threadIdx.x >> 5);
    int lane  = threadIdx.x & 31;
    if (group >= NGROUPS) return;

    int b   = group / (HDIM * GROUPS_PER_ROW);
    int rem = group - b * (HDIM * GROUPS_PER_ROW);
    int h0  = rem / GROUPS_PER_ROW;
    int w0  = (rem - h0 * GROUPS_PER_ROW) * 16;

    int col   = lane & 15;        // output pixel within group (B column N)
    int khalf = (lane >> 4) << 1; // lane group K offset: 0 or 2

    v8f c = {};
    #pragma unroll
    for (int kb = 0; kb < 100; kb += 4) {
        int k0 = kb + khalf; // even; pairs (k0: c=max, k0+1: c=avg)

        // A: weights, row M=0 only
        v2f av = {0.0f, 0.0f};
        if ((lane & 15) == 0) {
            av.x = wlds[k0];
            av.y = wlds[k0 + 1];
        }

        // B: pooled (max, avg) pair at tap (dy, dx) for pixel column `col`
        int dy = k0 / 14;
        int r  = k0 - dy * 14;
        int dx = r >> 1;
        v2f bv = {0.0f, 0.0f};
        int hh = h0 + dy - 3;
        int ww = w0 + col + dx - 3;
        if (k0 < 98 && (unsigned)hh < HDIM && (unsigned)ww < WDIM) {
            float2 p = pool[((size_t)b * HDIM + hh) * WDIM + ww];
            bv.x = p.x;
            bv.y = p.y;
        }

        c = __builtin_amdgcn_wmma_f32_16x16x4_f32(
                /*neg_a=*/false, av, /*neg_b=*/false, bv,
                /*c_mod=*/(short)0, c, /*reuse_a=*/false, /*reuse_b=*/false);
    }

    if (lane < 16) {
        float v = c[0] + conv_b[0]; // C row M=0, N=lane -> VGPR0, lanes 0-15
        attn[((size_t)b * HDIM + h0) * WDIM + w0 + lane] =
            1.0f / (1.0f + expf(-v));
    }
}

// ---------------------------------------------------------------------------
// Pass 3: out = x * attn (broadcast over 256 channels). float4 per thread.
// x/out streamed -> non-temporal; attn reused by 64 threads -> temporal.
// ---------------------------------------------------------------------------
__global__ void apply_kernel(const float* __restrict__ x,
                             const float* __restrict__ attn,
                             float* __restrict__ out) {
    int i = blockIdx.x * blockDim.x + threadIdx.x; // float4 index
    const v4f* x4   = (const v4f*)x;
    v4f*       out4 = (v4f*)out;

    float a = attn[i >> 6]; // 64 float4 per pixel
    v4f v = __builtin_nontemporal_load(x4 + i);
    v.x *= a; v.y *= a; v.z *= a; v.w *= a;
    __builtin_nontemporal_store(v, out4 + i);
}

extern "C" void kernel_launch(void* const* d_in, const int* in_sizes, int n_in,
                              void* d_out, int out_size, void* d_ws, size_t ws_size,
                              hipStream_t stream) {
    const float* x      = (const float*)d_in[0]; // [16,224,224,256]
    const float* conv_w = (const float*)d_in[1]; // [7,7,2,1] -> 98 floats
    const float* conv_b = (const float*)d_in[2]; // [1]
    float* out = (float*)d_out;

    // Workspace layout: pool float2[NPIX] | attn float[NPIX]
    float2* pool = (float2*)d_ws;
    float*  attn = (float*)((char*)d_ws + (size_t)NPIX * sizeof(float2));

    // Pass 1: 1 wave per pixel, 8 waves (256 thr) per block
    {
        int blocks = (NPIX + 7) / 8; // 100352
        pool_kernel<<<blocks, 256, 0, stream>>>(x, pool);
    }
    // Pass 2: 1 wave per 16-pixel group, 4 waves (128 thr) per block
    {
        int blocks = (NGROUPS + 3) / 4; // 12544
        conv_wmma_kernel<<<blocks, 128, 0, stream>>>(pool, conv_w, conv_b, attn);
    }
    // Pass 3: one float4 per thread
    {
        int n4 = (int)(NELEM / 4);          // 51380224 (exactly divisible)
        int blocks = n4 / 256;              // 200704
        apply_kernel<<<blocks, 256, 0, stream>>>(x, attn, out);
    }
}